// DownSample2_63290638073924
// MI455X (gfx1250) — compile-verified
//
#include <hip/hip_runtime.h>
#include <hip/hip_bf16.h>
#include <math.h>

#define BS 16
#define EMB 512
#define N_STK 32
#define N_STK_PNT 64
#define SP_OUT 1024
#define DN_OUT 1024
#define NS2 16            /* N_STK/2 */
#define NP2 32            /* N_STK_PNT/2 */
#define KDN 1536          /* EMB*3 */
#define EPS 1e-5f
#define CK  32            /* channels staged in LDS per chunk */

typedef __attribute__((ext_vector_type(2))) float v2f;
typedef __attribute__((ext_vector_type(8))) float v8f;

__device__ __forceinline__ float gelu_exact(float v) {
    return 0.5f * v * (1.0f + erff(v * 0.70710678118654752440f));
}

// ---------------------------------------------------------------------------
// K1: FPS (16 samples, start idx 0) + 2-NN per stroke, select at fps indices.
// One wave (32 lanes) per batch; lane == stroke index.
// ---------------------------------------------------------------------------
__global__ void fps_knn_kernel(const float* __restrict__ stk,
                               int* __restrict__ fps_idx,
                               int* __restrict__ knn_sel) {
    const int b = blockIdx.x;
    const int lane = threadIdx.x;                 // 0..31
    const float* X = stk + (size_t)b * EMB * N_STK;   // [512][32]

    __shared__ float dist_sh[N_STK];
    __shared__ int   far_sh;
    __shared__ int   fidx[NS2];
    __shared__ int   k1_sh[N_STK], k2_sh[N_STK];

    // ---- farthest point sampling ----
    float dist = 1e10f;
    int far = 0;
    for (int t = 0; t < NS2; ++t) {
        if (lane == 0) fidx[t] = far;
        float d = 0.f;
        for (int c = 0; c < EMB; ++c) {
            float diff = X[c * N_STK + lane] - X[c * N_STK + far];
            d += diff * diff;
        }
        dist = fminf(dist, d);
        dist_sh[lane] = dist;
        __syncthreads();
        if (lane == 0) {                           // argmax, first-max wins
            float best = dist_sh[0]; int bi = 0;
            for (int m = 1; m < N_STK; ++m)
                if (dist_sh[m] > best) { best = dist_sh[m]; bi = m; }
            far_sh = bi;
        }
        __syncthreads();
        far = far_sh;
    }

    // ---- 2-NN (incl. self): two smallest squared distances, ties -> lowest idx
    float best1 = 3.4e38f, best2 = 3.4e38f;
    int i1 = 0, i2 = 0;
    for (int m = 0; m < N_STK; ++m) {
        float d = 0.f;
        for (int c = 0; c < EMB; ++c) {
            float diff = X[c * N_STK + lane] - X[c * N_STK + m];
            d += diff * diff;
        }
        if (d < best1)      { best2 = best1; i2 = i1; best1 = d; i1 = m; }
        else if (d < best2) { best2 = d; i2 = m; }
    }
    k1_sh[lane] = i1;
    k2_sh[lane] = i2;
    __syncthreads();

    if (lane < NS2) {
        int r = fidx[lane];
        fps_idx[b * NS2 + lane] = r;
        knn_sel[(b * NS2 + lane) * 2 + 0] = k1_sh[r];
        knn_sel[(b * NS2 + lane) * 2 + 1] = k2_sh[r];
    }
}

// ---------------------------------------------------------------------------
// K2: stk_sampled[b][c][i] = stk_fea[b][c][fps_idx[b][i]]
// ---------------------------------------------------------------------------
__global__ void gather_stk_kernel(const float* __restrict__ stk,
                                  const int* __restrict__ fps_idx,
                                  float* __restrict__ out) {
    int idx = blockIdx.x * blockDim.x + threadIdx.x;   // BS*EMB*NS2
    int i = idx & (NS2 - 1);
    int c = (idx >> 4) & (EMB - 1);
    int b = idx >> 13;
    int n = fps_idx[b * NS2 + i];
    out[idx] = stk[((size_t)b * EMB + c) * N_STK + n];
}

// ---------------------------------------------------------------------------
// K3: sfin[b][c][i] = max over the 2-NN group of sparse_fea columns
// ---------------------------------------------------------------------------
__global__ void sparse_prep_kernel(const float* __restrict__ sp,
                                   const int* __restrict__ knn_sel,
                                   float* __restrict__ sfin) {
    int idx = blockIdx.x * blockDim.x + threadIdx.x;   // BS*EMB*NS2
    int i = idx & (NS2 - 1);
    int c = (idx >> 4) & (EMB - 1);
    int b = idx >> 13;
    int k0 = knn_sel[(b * NS2 + i) * 2 + 0];
    int k1 = knn_sel[(b * NS2 + i) * 2 + 1];
    const float* row = sp + ((size_t)b * EMB + c) * N_STK;
    sfin[idx] = fmaxf(row[k0], row[k1]);
}

// ---------------------------------------------------------------------------
// K4: sparse GEMM  sf[b][o][i] = gelu(bn(sp_w[o,:] . sfin[b,:,i] + sp_b[o]))
// Per-wave 16x16 tile, K=512 via V_WMMA_F32_16X16X4_F32.
// ---------------------------------------------------------------------------
__global__ void sparse_gemm_kernel(const float* __restrict__ W,
                                   const float* __restrict__ Bm,
                                   const float* __restrict__ bias,
                                   const float* __restrict__ g,
                                   const float* __restrict__ be,
                                   const float* __restrict__ mean,
                                   const float* __restrict__ var,
                                   float* __restrict__ out) {
    int wid  = blockIdx.x;               // 0 .. BS*64-1
    int b    = wid >> 6;
    int m0   = (wid & 63) * 16;
    int lane = threadIdx.x;
    int half = lane >> 4;                // K split
    int col  = lane & 15;                // M row for A, N col for B/D
    const float* Bp = Bm + (size_t)b * EMB * NS2;

    v8f acc = {};
    for (int k = 0; k < EMB; k += 4) {
        int kk = k + 2 * half;
        v2f a, bb;
        const float* wrow = W + (size_t)(m0 + col) * EMB;
        a.x  = wrow[kk];
        a.y  = wrow[kk + 1];
        bb.x = Bp[kk * NS2 + col];
        bb.y = Bp[(kk + 1) * NS2 + col];
        acc = __builtin_amdgcn_wmma_f32_16x16x4_f32(
            false, a, false, bb, (short)0, acc, false, false);
    }
#pragma unroll
    for (int j = 0; j < 8; ++j) {
        int o = m0 + j + 8 * half;
        float v  = acc[j] + bias[o];
        float sc = g[o] * rsqrtf(var[o] + EPS);
        v = sc * (v - mean[o]) + be[o];
        out[((size_t)b * SP_OUT + o) * NS2 + col] = gelu_exact(v);
    }
}

// ---------------------------------------------------------------------------
// K5: dense conv as 3 accumulated GEMMs (one per kx tap), K=512 each.
// Block = 4 waves, owns one (b,s) pair and 64 output rows.  Input slice
// X[b, c0..c0+31, s, 0..63] is staged to LDS with global_load_async_to_lds
// (ASYNCcnt); each wave computes a 16(M) x 32(N) region with two
// accumulators sharing one A fragment per WMMA step.
// ---------------------------------------------------------------------------
__global__ void __launch_bounds__(128)
dense_gemm_kernel(const float* __restrict__ W,
                  const float* __restrict__ X,
                  const float* __restrict__ bias,
                  const float* __restrict__ g,
                  const float* __restrict__ be,
                  const float* __restrict__ mean,
                  const float* __restrict__ var,
                  float* __restrict__ conv) {
    __shared__ float tile[CK * N_STK_PNT];       // 32 x 64 floats = 8 KB

    int pair = blockIdx.x >> 4;                  // 0..511 : (b, s)
    int mblk = blockIdx.x & 15;                  // 16 M-blocks of 64 rows
    int b    = pair >> 5;
    int s    = pair & 31;
    int wave = threadIdx.x >> 5;                 // 0..3
    int lane = threadIdx.x & 31;
    int half = lane >> 4;
    int col  = lane & 15;
    int m0   = mblk * 64 + wave * 16;

    const float* Xp  = X + ((size_t)b * EMB) * (N_STK * N_STK_PNT) + s * N_STK_PNT;
    const float* wr0 = W + (size_t)(m0 + col) * KDN;
    unsigned lbase = (unsigned)(size_t)(void*)tile;

    v8f accLo = {};                              // columns p = 0..15
    v8f accHi = {};                              // columns p = 16..31

    for (int c0 = 0; c0 < EMB; c0 += CK) {
        // ---- stage chunk: 2048 floats via 4 async b128 per thread-slot ----
        __syncthreads();                         // protect LDS reuse
#pragma unroll
        for (int j = 0; j < 4; ++j) {
            int f4 = j * 128 + threadIdx.x;      // unit of 4 floats, 0..511
            int cc = f4 >> 4;                    // row in chunk (0..31)
            int xi = (f4 & 15) * 4;              // col (0..60)
            unsigned long long ga =
                (unsigned long long)(const void*)(Xp + (size_t)(c0 + cc) * (N_STK * N_STK_PNT) + xi);
            unsigned ldsoff = lbase + (unsigned)((cc * N_STK_PNT + xi) * 4);
            asm volatile("global_load_async_to_lds_b128 %0, %1, off"
                         :: "v"(ldsoff), "v"(ga) : "memory");
        }
        asm volatile("s_wait_asynccnt 0x0" ::: "memory");
        __syncthreads();

        // ---- compute: 3 taps x 8 k-steps x 2 wmma ----
#pragma unroll
        for (int kx = 0; kx < 3; ++kx) {
            int   xiLo = 2 * col - 1 + kx;       // -1..31
            int   xiHi = xiLo + 32;              // 31..63
            float mLo  = (xiLo >= 0) ? 1.f : 0.f;
            int   xLo  = xiLo < 0 ? 0 : xiLo;
            for (int c4 = 0; c4 < CK; c4 += 4) {
                int cc = c4 + 2 * half;
                v2f a, bLo, bHi;
                const float* wr = wr0 + (size_t)(c0 + cc) * 3 + kx;
                a.x   = wr[0];
                a.y   = wr[3];
                bLo.x = mLo * tile[cc * N_STK_PNT + xLo];
                bLo.y = mLo * tile[(cc + 1) * N_STK_PNT + xLo];
                bHi.x = tile[cc * N_STK_PNT + xiHi];
                bHi.y = tile[(cc + 1) * N_STK_PNT + xiHi];
                accLo = __builtin_amdgcn_wmma_f32_16x16x4_f32(
                    false, a, false, bLo, (short)0, accLo, false, false);
                accHi = __builtin_amdgcn_wmma_f32_16x16x4_f32(
                    false, a, false, bHi, (short)0, accHi, false, false);
            }
        }
    }

    // ---- epilogue: bias + BN + GELU ----
#pragma unroll
    for (int j = 0; j < 8; ++j) {
        int o = m0 + j + 8 * half;
        float sc = g[o] * rsqrtf(var[o] + EPS);
        float vL = sc * (accLo[j] + bias[o] - mean[o]) + be[o];
        float vH = sc * (accHi[j] + bias[o] - mean[o]) + be[o];
        float* dst = conv + (((size_t)b * DN_OUT + o) * N_STK + s) * NP2;
        dst[col]      = gelu_exact(vL);
        dst[col + 16] = gelu_exact(vH);
    }
}

// ---------------------------------------------------------------------------
// K6: df[b][o][i*32+p] = max(conv[b][o][k0][p], conv[b][o][k1][p])
// ---------------------------------------------------------------------------
__global__ void dense_gather_kernel(const float* __restrict__ conv,
                                    const int* __restrict__ knn_sel,
                                    float* __restrict__ df) {
    int idx = blockIdx.x * blockDim.x + threadIdx.x;   // BS*DN_OUT*NS2*NP2
    int p = idx & (NP2 - 1);
    int i = (idx >> 5) & (NS2 - 1);
    int o = (idx >> 9) & (DN_OUT - 1);
    int b = idx >> 19;
    int k0 = knn_sel[(b * NS2 + i) * 2 + 0];
    int k1 = knn_sel[(b * NS2 + i) * 2 + 1];
    const float* base = conv + ((size_t)b * DN_OUT + o) * N_STK * NP2;
    df[idx] = fmaxf(base[k0 * NP2 + p], base[k1 * NP2 + p]);
}

// ---------------------------------------------------------------------------
extern "C" void kernel_launch(void* const* d_in, const int* in_sizes, int n_in,
                              void* d_out, int out_size, void* d_ws, size_t ws_size,
                              hipStream_t stream) {
    const float* sparse = (const float*)d_in[0];
    const float* dense  = (const float*)d_in[1];
    const float* stk    = (const float*)d_in[2];
    const float* sp_w   = (const float*)d_in[3];
    const float* sp_b   = (const float*)d_in[4];
    const float* sp_g   = (const float*)d_in[5];
    const float* sp_be  = (const float*)d_in[6];
    const float* sp_m   = (const float*)d_in[7];
    const float* sp_v   = (const float*)d_in[8];
    const float* dn_w   = (const float*)d_in[9];
    const float* dn_b   = (const float*)d_in[10];
    const float* dn_g   = (const float*)d_in[11];
    const float* dn_be  = (const float*)d_in[12];
    const float* dn_m   = (const float*)d_in[13];
    const float* dn_v   = (const float*)d_in[14];

    float* out      = (float*)d_out;
    float* sf_out   = out;                                      // [16,1024,16]
    float* df_out   = out + (size_t)BS * SP_OUT * NS2;          // [16,1024,512]
    float* stk_out  = df_out + (size_t)BS * DN_OUT * NS2 * NP2; // [16,512,16]

    char*  ws   = (char*)d_ws;
    int*   fpsi = (int*)ws;                  // 256 ints
    int*   knns = (int*)(ws + 1024);         // 512 ints
    float* sfin = (float*)(ws + 4096);       // 131072 floats
    float* conv = (float*)(ws + (1 << 20));  // 16M floats (64 MB)

    fps_knn_kernel<<<BS, 32, 0, stream>>>(stk, fpsi, knns);
    gather_stk_kernel<<<(BS * EMB * NS2) / 256, 256, 0, stream>>>(stk, fpsi, stk_out);
    sparse_prep_kernel<<<(BS * EMB * NS2) / 256, 256, 0, stream>>>(sparse, knns, sfin);
    sparse_gemm_kernel<<<BS * (SP_OUT / 16), 32, 0, stream>>>(
        sp_w, sfin, sp_b, sp_g, sp_be, sp_m, sp_v, sf_out);
    dense_gemm_kernel<<<512 * 16, 128, 0, stream>>>(
        dn_w, dense, dn_b, dn_g, dn_be, dn_m, dn_v, conv);
    dense_gather_kernel<<<(BS * DN_OUT * NS2 * NP2) / 256, 256, 0, stream>>>(
        conv, knns, df_out);
}